// PostSpectralContext_32375463477504
// MI455X (gfx1250) — compile-verified
//
#include <hip/hip_runtime.h>
#include <hip/hip_bf16.h>

#define N_BOXES     512
#define NUM_CLASSES 151
#define IN_DIM      4096
#define NMS_THRESH  0.3f

typedef __attribute__((ext_vector_type(2))) float v2f;
typedef __attribute__((ext_vector_type(8))) float v8f;

// ---------------------------------------------------------------------------
// GEMM: obj_dists[512,151] = obj_fmap[512,4096] @ W[4096,151] + b
// One wave per 16x16 output tile using V_WMMA_F32_16X16X4_F32.
// A operand (16x4):  lanes 0-15 hold K=k+0/k+1, lanes 16-31 hold K=k+2/k+3.
// B operand (4x16):  mirrored; per-lane column N = lane&15.
// C/D (16x16 f32):   VGPR r -> row = r + (lane>>4)*8, col = lane&15.
//
// Edge handling: B column n only contributes to output column n, and we never
// store columns >= 151 — so edge lanes just CLAMP their column address (values
// are don't-cares). No predication in the K-loop -> clean load clauses + WMMA.
// ---------------------------------------------------------------------------
__global__ __launch_bounds__(32)
void gemm_wmma_f32(const float* __restrict__ A, const float* __restrict__ W,
                   const float* __restrict__ bias, float* __restrict__ out)
{
  const int tile = blockIdx.x;            // 32 M-tiles x 10 N-tiles
  const int mt   = tile / 10;
  const int nt   = tile - mt * 10;
  const int lane = threadIdx.x & 31;
  const int hi   = lane >> 4;             // 0 or 1 (K half)
  const int l15  = lane & 15;
  const int arow = mt * 16 + l15;
  const int bcol = nt * 16 + l15;
  const bool bok = (bcol < NUM_CLASSES);
  const int  bc  = bok ? bcol : (NUM_CLASSES - 1);   // clamped, loop-invariant

  const float* __restrict__ Ar = A + (size_t)arow * IN_DIM + hi * 2;
  const float* __restrict__ Bp = W + bc + (size_t)hi * 2 * NUM_CLASSES;
  v8f acc = {};

#pragma unroll 8
  for (int k = 0; k < IN_DIM; k += 4) {
    v2f a, b;
    a.x = Ar[k];
    a.y = Ar[k + 1];
    b.x = Bp[(size_t)k * NUM_CLASSES];
    b.y = Bp[(size_t)(k + 1) * NUM_CLASSES];
    acc = __builtin_amdgcn_wmma_f32_16x16x4_f32(
        /*neg_a=*/false, a, /*neg_b=*/false, b,
        /*c_mod=*/(short)0, acc, /*reuse_a=*/false, /*reuse_b=*/false);
  }

  if (bok) {
    const float bb = bias[bcol];
#pragma unroll
    for (int r = 0; r < 8; ++r) {
      const int row = mt * 16 + hi * 8 + r;
      out[(size_t)row * NUM_CLASSES + bcol] = acc[r] + bb;
    }
  }
}

// Orderable key: higher score wins; on equal score, LOWER flat index wins
// (matches jnp.argmax tie-breaking). Scores here are in {-1} U [0,1].
__device__ __forceinline__ unsigned long long packKey(float v, int flat) {
  unsigned int b = __float_as_uint(v);
  b = (b & 0x80000000u) ? ~b : (b | 0x80000000u);   // monotonic float->uint
  return ((unsigned long long)b << 32) |
         (unsigned long long)(0x7FFFFFFFu - (unsigned int)flat);
}
__device__ __forceinline__ int keyFlat(unsigned long long k) {
  return (int)(0x7FFFFFFFu - (unsigned int)(k & 0xFFFFFFFFull));
}

// ---------------------------------------------------------------------------
// Greedy NMS decode. Single workgroup; full 512x151 score matrix lives in the
// 320KB WGP LDS. Incremental per-row argmax keys; ds_max_u64 atomics + wave32
// shuffles for the global argmax each step.
// ---------------------------------------------------------------------------
__global__ __launch_bounds__(1024)
void greedy_nms_decode_kernel(const float* __restrict__ dists,
                              const float* __restrict__ boxes,
                              float* __restrict__ preds_out)
{
  __shared__ float              s_scores[N_BOXES * NUM_CLASSES]; // 302 KB
  __shared__ unsigned long long s_rowKey[N_BOXES];               // 4 KB
  __shared__ unsigned long long s_best;

  const int tid = threadIdx.x;

  // ---- init: row-wise softmax, zero background, build per-row keys --------
  if (tid < N_BOXES) {
    const float* row  = dists + (size_t)tid * NUM_CLASSES;
    float*       srow = s_scores + tid * NUM_CLASSES;
    float mx = row[0];
    for (int c = 1; c < NUM_CLASSES; ++c) mx = fmaxf(mx, row[c]);
    float sum = 0.0f;
    for (int c = 0; c < NUM_CLASSES; ++c) {
      const float e = __expf(row[c] - mx);
      srow[c] = e;
      sum += e;
    }
    const float inv = 1.0f / sum;
    unsigned long long best = 0ull;
    for (int c = 0; c < NUM_CLASSES; ++c) {
      const float v = (c == 0) ? 0.0f : srow[c] * inv;
      srow[c] = v;
      const unsigned long long kk = packKey(v, tid * NUM_CLASSES + c);
      if (kk > best) best = kk;
    }
    s_rowKey[tid]  = best;
    preds_out[tid] = 0.0f;                 // commits initialized to zero
  }
  if (tid == 0) s_best = 0ull;
  __syncthreads();

  // ---- 512 sequential greedy steps ----------------------------------------
  for (int step = 0; step < N_BOXES; ++step) {
    // global argmax = max over 512 row keys
    unsigned long long k = (tid < N_BOXES) ? s_rowKey[tid] : 0ull;
    for (int m = 16; m > 0; m >>= 1) {
      const unsigned long long o = __shfl_xor(k, m, 32);
      if (o > k) k = o;
    }
    if (tid < N_BOXES && (tid & 31) == 0) atomicMax(&s_best, k);
    __syncthreads();
    const unsigned long long best = s_best;
    __syncthreads();                        // all read before reset

    const int flat = keyFlat(best);
    const int box  = flat / NUM_CLASSES;
    const int cls  = flat - box * NUM_CLASSES;

    if (tid == 0) {
      s_best = 0ull;
      preds_out[box] = (float)cls;          // commit
    }

    // chosen box geometry (uniform load, all lanes same address)
    const float* cb  = boxes + ((size_t)box * NUM_CLASSES + cls) * 4;
    const float cx1 = cb[0], cy1 = cb[1], cx2 = cb[2], cy2 = cb[3];
    const float areaC = (cx2 - cx1 + 1.0f) * (cy2 - cy1 + 1.0f);

    // zero class-`cls` column where IoU >= thresh (rows other than `box`)
    if (tid < N_BOXES && tid != box) {
      const int   n   = tid;
      const int   fl  = n * NUM_CLASSES + cls;
      const float old = s_scores[fl];
      if (old != 0.0f) {
        const float* nb = boxes + ((size_t)n * NUM_CLASSES + cls) * 4;
        const float nx1 = nb[0], ny1 = nb[1], nx2 = nb[2], ny2 = nb[3];
        const float areaN = (nx2 - nx1 + 1.0f) * (ny2 - ny1 + 1.0f);
        const float iw = fmaxf(fminf(cx2, nx2) - fmaxf(cx1, nx1) + 1.0f, 0.0f);
        const float ih = fmaxf(fminf(cy2, ny2) - fmaxf(cy1, ny1) + 1.0f, 0.0f);
        const float inter = iw * ih;
        const float iou   = inter / (areaC + areaN - inter);
        if (iou >= NMS_THRESH) {
          s_scores[fl] = 0.0f;
          const unsigned long long nk = packKey(0.0f, fl);
          const unsigned long long rk = s_rowKey[n];
          if (nk > rk) {
            s_rowKey[n] = nk;               // retired row resurrected to 0
          } else if (keyFlat(rk) == fl) {
            // invalidated the row's best -> rescan this row
            unsigned long long bb = 0ull;
            const float* sr = s_scores + n * NUM_CLASSES;
            for (int c = 0; c < NUM_CLASSES; ++c) {
              const unsigned long long kk = packKey(sr[c], n * NUM_CLASSES + c);
              if (kk > bb) bb = kk;
            }
            s_rowKey[n] = bb;
          }
        }
      }
    }

    // retire chosen row (disjoint addresses from the column update above)
    if (tid >= 512 && tid < 512 + NUM_CLASSES)
      s_scores[box * NUM_CLASSES + (tid - 512)] = -1.0f;
    if (tid == 512 + NUM_CLASSES)
      s_rowKey[box] = packKey(-1.0f, box * NUM_CLASSES);

    __syncthreads();
  }
}

extern "C" void kernel_launch(void* const* d_in, const int* in_sizes, int n_in,
                              void* d_out, int out_size, void* d_ws, size_t ws_size,
                              hipStream_t stream) {
  (void)in_sizes; (void)n_in; (void)out_size; (void)d_ws; (void)ws_size;
  const float* obj_fmap = (const float*)d_in[0];   // [512,4096]
  const float* boxes    = (const float*)d_in[1];   // [512,151,4]
  const float* W        = (const float*)d_in[2];   // [4096,151]
  const float* bias     = (const float*)d_in[3];   // [151]

  float* obj_dists = (float*)d_out;                        // [512,151]
  float* preds     = obj_dists + N_BOXES * NUM_CLASSES;    // [512] as float

  gemm_wmma_f32<<<320, 32, 0, stream>>>(obj_fmap, W, bias, obj_dists);
  greedy_nms_decode_kernel<<<1, 1024, 0, stream>>>(obj_dists, boxes, preds);
}